// GCN_79568564126323
// MI455X (gfx1250) — compile-verified
//
#include <hip/hip_runtime.h>

typedef float v2f __attribute__((ext_vector_type(2)));
typedef float v8f __attribute__((ext_vector_type(8)));

// Native f32 atomic add: relaxed, device (agent) scope -> GLOBAL_ATOMIC_ADD_F32
__device__ __forceinline__ void atomic_add_f32(float* p, float v) {
    __hip_atomic_fetch_add(p, v, __ATOMIC_RELAXED, __HIP_MEMORY_SCOPE_AGENT);
}

// XOR-butterfly sum over the 16 lanes of each wave32 half using DS_SWIZZLE
// group-of-32 mode: offset = xor_mask<<10 | or_mask<<5 | and_mask.
__device__ __forceinline__ float half16_sum(float v) {
    v += __int_as_float(__builtin_amdgcn_ds_swizzle(__float_as_int(v), 0x041F)); // xor 1
    v += __int_as_float(__builtin_amdgcn_ds_swizzle(__float_as_int(v), 0x081F)); // xor 2
    v += __int_as_float(__builtin_amdgcn_ds_swizzle(__float_as_int(v), 0x101F)); // xor 4
    v += __int_as_float(__builtin_amdgcn_ds_swizzle(__float_as_int(v), 0x201F)); // xor 8
    return v;
}

// ---------------------------------------------------------------------------
// K1: init deg (=1.0 for the self loop) and zero the 4-wide aggregation buffer
// ---------------------------------------------------------------------------
__global__ void gcn_init(float* __restrict__ deg, float4* __restrict__ agg, int n) {
    int i = blockIdx.x * blockDim.x + threadIdx.x;
    if (i < n) {
        deg[i] = 1.0f;                       // self loop counts once
        agg[i] = make_float4(0.f, 0.f, 0.f, 0.f);
    }
}

// ---------------------------------------------------------------------------
// K2: degree accumulation over edges (dst side)
// ---------------------------------------------------------------------------
__global__ void gcn_deg(const long long* __restrict__ dst, float* __restrict__ deg, int e) {
    int i = blockIdx.x * blockDim.x + threadIdx.x;
    if (i < e) atomic_add_f32(&deg[(int)dst[i]], 1.0f);
}

// ---------------------------------------------------------------------------
// K3: deg -> dinv in place (deg >= 1 always, so plain rsqrt)
// ---------------------------------------------------------------------------
__global__ void gcn_rsqrt(float* __restrict__ dinv, int n) {
    int i = blockIdx.x * blockDim.x + threadIdx.x;
    if (i < n) dinv[i] = rsqrtf(dinv[i]);
}

// ---------------------------------------------------------------------------
// K4: layer-1 aggregation of the RAW 4-wide features:
//     agg[d] += dinv[s]*dinv[d] * x[s]          (4 f32 atomics per edge)
// ---------------------------------------------------------------------------
__global__ void gcn_agg(const long long* __restrict__ src, const long long* __restrict__ dst,
                        const float* __restrict__ dinv, const float4* __restrict__ x4,
                        float* __restrict__ agg, int e) {
    int i = blockIdx.x * blockDim.x + threadIdx.x;
    if (i >= e) return;
    int s = (int)src[i];
    int d = (int)dst[i];
    float nrm = dinv[s] * dinv[d];
    float4 xs = x4[s];
    float* a = agg + 4 * d;
    atomic_add_f32(a + 0, nrm * xs.x);
    atomic_add_f32(a + 1, nrm * xs.y);
    atomic_add_f32(a + 2, nrm * xs.z);
    atomic_add_f32(a + 3, nrm * xs.w);
}

// ---------------------------------------------------------------------------
// K5: dense transform per 16-node tile using V_WMMA_F32_16X16X4_F32:
//     v[i]   = agg[i] + dinv[i]^2 * x[i]              (fold self loop, [16x4])
//     h      = relu(v @ W1 + b1)                      ([16x16], in WMMA acc)
//     g[i]   = h @ W2                                 (ds_swizzle reduction)
// One tile per wave32. EXEC is all-ones for executing waves (uniform guard).
// ---------------------------------------------------------------------------
__global__ void gcn_dense_wmma(const float* __restrict__ x, const float* __restrict__ agg,
                               const float* __restrict__ dinv,
                               const float* __restrict__ W1, const float* __restrict__ b1,
                               const float* __restrict__ W2,
                               float* __restrict__ g, int n, int ntiles) {
    int lane = threadIdx.x & 31;
    int wave = threadIdx.x >> 5;
    int tile = blockIdx.x * (blockDim.x >> 5) + wave;
    if (tile >= ntiles) return;               // wave-uniform: EXEC stays all-1s

    int m    = lane & 15;                     // A: row / B,C,D: column index
    int half = lane >> 4;                     // 0: K=0..1 | 1: K=2..3
    int k0   = half << 1;

    int node = tile * 16 + m;
    if (node >= n) node = n - 1;              // clamp tail reads (N%16==0 here)

    float di = dinv[node];
    float sl = di * di;                       // self-loop norm dinv^2

    // A matrix: v = agg + dinv^2 * x, components (k0, k0+1) for this lane
    const float* xp = x   + node * 4 + k0;
    const float* ap = agg + node * 4 + k0;
    v2f a;
    a.x = ap[0] + sl * xp[0];
    a.y = ap[1] + sl * xp[1];

    // B matrix: W1[4][16] row-major; rows (k0, k0+1), column m
    v2f b;
    b.x = W1[(k0 + 0) * 16 + m];
    b.y = W1[(k0 + 1) * 16 + m];

    v8f c = {};
    c = __builtin_amdgcn_wmma_f32_16x16x4_f32(
            /*neg_a=*/false, a, /*neg_b=*/false, b,
            /*c_mod=*/(short)0, c, /*reuse_a=*/false, /*reuse_b=*/false);

    // This lane holds h[row r + 8*half][col m] in c[r]. Apply bias+ReLU, then
    // multiply by W2[m] and reduce over the 16 columns (lanes within a half).
    float bias = b1[m];
    float w2   = W2[m];
    float t[8];
#pragma unroll
    for (int r = 0; r < 8; ++r) {
        float h = c[r] + bias;
        h = h > 0.f ? h : 0.f;
        t[r] = half16_sum(h * w2);
    }

    if (m == 0) {                             // lanes 0 and 16 write 8 rows each
        int base = tile * 16 + half * 8;
#pragma unroll
        for (int r = 0; r < 8; ++r) {
            int idx = base + r;
            if (idx < n) g[idx] = t[r];
        }
    }
}

// ---------------------------------------------------------------------------
// K6: initialize output with layer-2 self-loop term: out[i] = dinv^2*g[i] + b2
// ---------------------------------------------------------------------------
__global__ void gcn_selfout(const float* __restrict__ dinv, const float* __restrict__ g,
                            const float* __restrict__ b2, float* __restrict__ out, int n) {
    int i = blockIdx.x * blockDim.x + threadIdx.x;
    if (i < n) out[i] = dinv[i] * dinv[i] * g[i] + b2[0];
}

// ---------------------------------------------------------------------------
// K7: layer-2 scalar scatter: out[d] += dinv[s]*dinv[d]*g[s]  (1 atomic/edge)
// ---------------------------------------------------------------------------
__global__ void gcn_scatter2(const long long* __restrict__ src, const long long* __restrict__ dst,
                             const float* __restrict__ dinv, const float* __restrict__ g,
                             float* __restrict__ out, int e) {
    int i = blockIdx.x * blockDim.x + threadIdx.x;
    if (i >= e) return;
    int s = (int)src[i];
    int d = (int)dst[i];
    atomic_add_f32(&out[d], dinv[s] * dinv[d] * g[s]);
}

extern "C" void kernel_launch(void* const* d_in, const int* in_sizes, int n_in,
                              void* d_out, int out_size, void* d_ws, size_t ws_size,
                              hipStream_t stream) {
    const float*     x  = (const float*)d_in[0];
    const long long* ei = (const long long*)d_in[1];
    const float*     W1 = (const float*)d_in[2];
    const float*     b1 = (const float*)d_in[3];
    const float*     W2 = (const float*)d_in[4];
    const float*     b2 = (const float*)d_in[5];
    float*           out = (float*)d_out;

    const int n = in_sizes[0] / 4;    // 200000 nodes
    const int e = in_sizes[1] / 2;    // 3200000 edges
    const long long* src = ei;        // edge_index[0]
    const long long* dst = ei + e;    // edge_index[1]

    // Workspace layout (floats): dinv[n] | agg[4n] | g[n]  => 6n*4 = 4.8 MB
    float* ws   = (float*)d_ws;
    float* dinv = ws;
    float* agg  = ws + (size_t)n;
    float* g    = ws + (size_t)5 * n;

    const int tb = 256;
    const int nb_n = (n + tb - 1) / tb;
    const int nb_e = (e + tb - 1) / tb;

    gcn_init  <<<nb_n, tb, 0, stream>>>(dinv, (float4*)agg, n);
    gcn_deg   <<<nb_e, tb, 0, stream>>>(dst, dinv, e);
    gcn_rsqrt <<<nb_n, tb, 0, stream>>>(dinv, n);
    gcn_agg   <<<nb_e, tb, 0, stream>>>(src, dst, dinv, (const float4*)x, agg, e);

    const int ntiles = (n + 15) / 16;
    const int wavesPerBlock = tb / 32;
    const int nb_t = (ntiles + wavesPerBlock - 1) / wavesPerBlock;
    gcn_dense_wmma<<<nb_t, tb, 0, stream>>>(x, agg, dinv, W1, b1, W2, g, n, ntiles);

    gcn_selfout <<<nb_n, tb, 0, stream>>>(dinv, g, b2, out, n);
    gcn_scatter2<<<nb_e, tb, 0, stream>>>(src, dst, dinv, g, out, e);
}